// DMIGNN_33148557591125
// MI455X (gfx1250) — compile-verified
//
#include <hip/hip_runtime.h>
#include <math.h>

#define B_ 256
#define L_ 50
#define D_ 128
#define K_ 4
#define V_ 50000
#define NV (V_ - 1)      // 49999
#define BL (B_ * L_)     // 12800
#define LENGTH_ 5.0f
#define BETA_ 0.01f
#define EPS_ 1e-12f

typedef float v2f __attribute__((ext_vector_type(2)));
typedef float v8f __attribute__((ext_vector_type(8)));

__device__ __forceinline__ v8f wmma4(v2f a, v2f b, v8f c) {
    // V_WMMA_F32_16X16X4_F32 : (neg_a, A, neg_b, B, c_mod, C, reuse_a, reuse_b)
    return __builtin_amdgcn_wmma_f32_16x16x4_f32(false, a, false, b, (short)0, c,
                                                 false, false);
}

__device__ __forceinline__ float sigmoidf_(float x) { return 1.0f / (1.0f + expf(-x)); }

// ---------------------------------------------------------------- hs (masked mean)
__global__ void hs_kernel(const float* __restrict__ hidden, const int* __restrict__ mask,
                          float* __restrict__ hs) {
    int b = blockIdx.x, d = threadIdx.x;
    float s = 0.f, ms = 0.f;
    for (int l = 0; l < L_; ++l) {
        float m = (float)mask[b * L_ + l];
        s += hidden[(size_t)(b * L_ + l) * D_ + d] * m;
        ms += m;
    }
    hs[b * D_ + d] = s / ms;
}

// ---------------------------------------------------------------- hs @ att_w1 (tiny)
__global__ void hsw1_kernel(const float* __restrict__ hs, const float* __restrict__ att_w1,
                            float* __restrict__ out) {
    __shared__ float row[D_];
    int b = blockIdx.x, d = threadIdx.x;
    row[d] = hs[b * D_ + d];
    __syncthreads();
    float acc = 0.f;
    for (int j = 0; j < D_; ++j) acc += row[j] * att_w1[j * D_ + d];
    out[b * D_ + d] = acc;
}

// ---------------------------------------------------------------- generic WMMA f32 GEMM
// AMODE: 0 = row-major A pointer, 1 = CAT(pos_emb[l] , hidden[row])
// EPI:   0 = tanh, 1 = sigmoid(x + bias[col]), 2 = tanh(x + rowadd[(row/L)*D + col])
template <int AMODE, int EPI>
__global__ void gemm16_kernel(const float* __restrict__ A, const float* __restrict__ Bm,
                              float* __restrict__ C, const float* __restrict__ bias,
                              const float* __restrict__ rowadd,
                              const float* __restrict__ pos, const float* __restrict__ hid,
                              int M, int N, int Kd, int lda, int ldb, int ldc) {
    int wave = threadIdx.x >> 5;
    int lane = threadIdx.x & 31;
    int ntn = N >> 4;
    int tile = blockIdx.x * (blockDim.x >> 5) + wave;
    int mt = tile / ntn, nt = tile % ntn;
    if (mt >= (M >> 4)) return;

    int half = lane >> 4, q = lane & 15;
    int m = mt * 16 + q;
    int n = nt * 16 + q;

    v8f acc = {0.f, 0.f, 0.f, 0.f, 0.f, 0.f, 0.f, 0.f};
    for (int k0 = 0; k0 < Kd; k0 += 4) {
        int ka = k0 + 2 * half;
        v2f a;
        if (AMODE == 0) {
            a = *(const v2f*)(A + (size_t)m * lda + ka);
        } else {
            int l = m % L_;
            if (ka < D_) a = *(const v2f*)(pos + (size_t)l * D_ + ka);
            else         a = *(const v2f*)(hid + (size_t)m * D_ + (ka - D_));
        }
        v2f bf;  // NN layout: two strided b32 loads
        bf.x = Bm[(size_t)ka * ldb + n];
        bf.y = Bm[(size_t)(ka + 1) * ldb + n];
        acc = wmma4(a, bf, acc);
    }
    int col = nt * 16 + q;
    for (int r = 0; r < 8; ++r) {
        int row = mt * 16 + r + 8 * half;
        float v = acc[r];
        if (EPI == 0) v = tanhf(v);
        else if (EPI == 1) v = sigmoidf_(v + bias[col]);
        else               v = tanhf(v + rowadd[(size_t)(row / L_) * D_ + col]);
        C[(size_t)row * ldc + col] = v;
    }
}

// ---------------------------------------------------------------- beta + alpha gate
__global__ void beta_kernel(const float* __restrict__ g, const float* __restrict__ att,
                            const float* __restrict__ w2, const float* __restrict__ att_v,
                            const int* __restrict__ mask, float* __restrict__ beta) {
    int row = blockIdx.x;           // b*L + l
    int k = threadIdx.x >> 5;       // 0..3 (one wave per interest)
    int lane = threadIdx.x & 31;
    float s1 = 0.f, s2 = 0.f;
    for (int i = 0; i < 4; ++i) {
        int d = lane + 32 * i;
        s1 += g[(size_t)row * (K_ * D_) + k * D_ + d] * w2[d * K_ + k];
        s2 += att[(size_t)row * D_ + d] * att_v[d * K_ + k];
    }
    for (int off = 16; off > 0; off >>= 1) {
        s1 += __shfl_down(s1, off);
        s2 += __shfl_down(s2, off);
    }
    if (lane == 0) {
        float alpha = sigmoidf_(s2);
        beta[row * K_ + k] = s1 * (0.5f + alpha) * (float)mask[row];
    }
}

// ---------------------------------------------------------------- per-b normalize + Gram + sim
__global__ void sim_kernel(const float* __restrict__ beta, const int* __restrict__ mask,
                           float* __restrict__ sims) {
    __shared__ float nb[L_ * K_];
    __shared__ float nrm[K_];
    __shared__ float Gs[8];
    int b = blockIdx.x, t = threadIdx.x;  // 32 threads
    for (int i = t; i < L_ * K_; i += 32) nb[i] = beta[(size_t)b * L_ * K_ + i];
    __syncthreads();
    if (t < K_) {
        float s = 0.f;
        for (int l = 0; l < L_; ++l) { float v = nb[l * K_ + t]; s += v * v; }
        nrm[t] = fmaxf(sqrtf(s), EPS_);
    }
    __syncthreads();
    if (t < 6) {
        const int pi[6] = {0, 0, 0, 1, 1, 2};
        const int pj[6] = {1, 2, 3, 2, 3, 3};
        int i = pi[t], j = pj[t];
        float s = 0.f;
        for (int l = 0; l < L_; ++l)
            s += (nb[l * K_ + i] / nrm[i]) * (nb[l * K_ + j] / nrm[j]);
        Gs[t] = fabsf(s);
    }
    __syncthreads();
    if (t == 0) {
        float sim = 0.f;
        for (int p = 0; p < 6; ++p) sim += Gs[p];
        sim *= 2.0f / (K_ * (K_ - 1));
        float lens = 0.f;
        for (int l = 0; l < L_; ++l) lens += (float)mask[b * L_ + l];
        lens -= LENGTH_;
        sims[b] = sigmoidf_(sim * lens);
    }
}

// ---------------------------------------------------------------- fixed-order loss reduce
__global__ void loss_kernel(const float* __restrict__ sims, float* __restrict__ out_loss) {
    __shared__ float buf[B_];
    int t = threadIdx.x;
    buf[t] = sims[t];
    __syncthreads();
    for (int s = B_ / 2; s > 0; s >>= 1) {
        if (t < s) buf[t] += buf[t + s];
        __syncthreads();
    }
    if (t == 0) *out_loss = buf[0] * BETA_;
}

// ---------------------------------------------------------------- select[k,b,d]
__global__ void select_kernel(const float* __restrict__ beta, const float* __restrict__ hidden,
                              float* __restrict__ sel) {
    int kb = blockIdx.x;            // k*B + b
    int k = kb / B_, b = kb % B_;
    int d = threadIdx.x;
    float s = 0.f;
    for (int l = 0; l < L_; ++l)
        s += beta[(size_t)(b * L_ + l) * K_ + k] * hidden[(size_t)(b * L_ + l) * D_ + d];
    sel[(size_t)kb * D_ + d] = s;
}

// ---------------------------------------------------------------- normalize emb[1:]
__global__ void bnorm_kernel(const float* __restrict__ emb, float* __restrict__ bn) {
    __shared__ float part[4];
    int v = blockIdx.x;             // row v -> emb row v+1
    int d = threadIdx.x;            // 128
    float x = emb[(size_t)(v + 1) * D_ + d];
    float s = x * x;
    for (int off = 16; off > 0; off >>= 1) s += __shfl_down(s, off);
    if ((threadIdx.x & 31) == 0) part[threadIdx.x >> 5] = s;
    __syncthreads();
    float n = fmaxf(sqrtf(part[0] + part[1] + part[2] + part[3]), EPS_);
    bn[(size_t)v * D_ + d] = x / n;
}

// ---------------------------------------------------------------- big GEMM + fused max-over-K
// One wave owns a 16(b) x 16(v) tile for ALL 4 interests: 4 WMMA accumulators
// share each B-fragment of b_norm, then max fused in registers. Outputs are
// streamed with non-temporal stores (write-once, 256 MB) so b_norm/select
// stay resident in L2 across all 50,000 tiles.
__global__ void scores_kernel(const float* __restrict__ sel, const float* __restrict__ bn,
                              float* __restrict__ scores, float* __restrict__ maxsc) {
    const int NT = (NV + 15) / 16;  // 3125
    int wave = threadIdx.x >> 5;
    int lane = threadIdx.x & 31;
    int tile = blockIdx.x * (blockDim.x >> 5) + wave;
    int bt = tile / NT, nt = tile % NT;
    if (bt >= B_ / 16) return;

    int half = lane >> 4, q = lane & 15;
    int n = nt * 16 + q;
    int nclamp = (n < NV) ? n : (NV - 1);

    const float* a0 = sel + (size_t)(0 * B_ + bt * 16 + q) * D_;
    const float* a1 = sel + (size_t)(1 * B_ + bt * 16 + q) * D_;
    const float* a2 = sel + (size_t)(2 * B_ + bt * 16 + q) * D_;
    const float* a3 = sel + (size_t)(3 * B_ + bt * 16 + q) * D_;
    const float* bp = bn + (size_t)nclamp * D_;

    v8f c0 = {0.f,0.f,0.f,0.f,0.f,0.f,0.f,0.f};
    v8f c1 = c0, c2 = c0, c3 = c0;
    for (int k0 = 0; k0 < D_; k0 += 4) {
        int ka = k0 + 2 * half;
        v2f bf = *(const v2f*)(bp + ka);     // NT layout: contiguous b64
        c0 = wmma4(*(const v2f*)(a0 + ka), bf, c0);
        c1 = wmma4(*(const v2f*)(a1 + ka), bf, c1);
        c2 = wmma4(*(const v2f*)(a2 + ka), bf, c2);
        c3 = wmma4(*(const v2f*)(a3 + ka), bf, c3);
    }
    bool valid = (n < NV);
    for (int r = 0; r < 8; ++r) {
        int brow = bt * 16 + r + 8 * half;
        float v0 = c0[r], v1 = c1[r], v2 = c2[r], v3 = c3[r];
        if (valid) {
            // streaming (non-temporal) stores: th:TH_STORE_NT
            __builtin_nontemporal_store(v0, &scores[((size_t)0 * B_ + brow) * NV + n]);
            __builtin_nontemporal_store(v1, &scores[((size_t)1 * B_ + brow) * NV + n]);
            __builtin_nontemporal_store(v2, &scores[((size_t)2 * B_ + brow) * NV + n]);
            __builtin_nontemporal_store(v3, &scores[((size_t)3 * B_ + brow) * NV + n]);
            __builtin_nontemporal_store(fmaxf(fmaxf(v0, v1), fmaxf(v2, v3)),
                                        &maxsc[(size_t)brow * NV + n]);
        }
    }
}

// ---------------------------------------------------------------- launch
extern "C" void kernel_launch(void* const* d_in, const int* in_sizes, int n_in,
                              void* d_out, int out_size, void* d_ws, size_t ws_size,
                              hipStream_t stream) {
    const float* hidden  = (const float*)d_in[0];
    const int*   mask    = (const int*)d_in[1];
    const float* pos_emb = (const float*)d_in[2];
    const float* w1      = (const float*)d_in[3];
    const float* w2      = (const float*)d_in[4];
    const float* glu1_w  = (const float*)d_in[5];
    const float* glu1_b  = (const float*)d_in[6];
    const float* att_w1  = (const float*)d_in[7];
    const float* att_w2  = (const float*)d_in[8];
    const float* att_v   = (const float*)d_in[9];
    const float* emb     = (const float*)d_in[10];

    float* out = (float*)d_out;
    float* out_max   = out;                        // [B, NV]
    float* out_loss  = out + (size_t)B_ * NV;      // [1]
    float* out_score = out_loss + 1;               // [K, B, NV]

    float* ws = (float*)d_ws;
    size_t off = 0;
    float* w_hs     = ws + off; off += (size_t)B_ * D_;
    float* w_hsw1   = ws + off; off += (size_t)B_ * D_;
    float* w_tmp1   = ws + off; off += (size_t)BL * D_;
    float* w_g      = ws + off; off += (size_t)BL * K_ * D_;
    float* w_att    = ws + off; off += (size_t)BL * D_;
    float* w_beta   = ws + off; off += (size_t)BL * K_;
    float* w_sims   = ws + off; off += (size_t)B_;
    float* w_sel    = ws + off; off += (size_t)K_ * B_ * D_;
    float* w_bn     = ws + off; off += (size_t)NV * D_;
    (void)ws_size; (void)in_sizes; (void)n_in; (void)out_size;

    hs_kernel<<<B_, D_, 0, stream>>>(hidden, mask, w_hs);
    hsw1_kernel<<<B_, D_, 0, stream>>>(w_hs, att_w1, w_hsw1);

    // tmp1 = tanh(cat(pos, hidden) @ w1)      M=12800 N=128 Kd=256
    gemm16_kernel<1, 0><<<(BL / 16) * (D_ / 16) / 8, 256, 0, stream>>>(
        nullptr, w1, w_tmp1, nullptr, nullptr, pos_emb, hidden,
        BL, D_, 2 * D_, 0, D_, D_);

    // g = sigmoid(tmp1 @ glu1_w + glu1_b)     M=12800 N=512 Kd=128
    gemm16_kernel<0, 1><<<(BL / 16) * ((K_ * D_) / 16) / 8, 256, 0, stream>>>(
        w_tmp1, glu1_w, w_g, glu1_b, nullptr, nullptr, nullptr,
        BL, K_ * D_, D_, D_, K_ * D_, K_ * D_);

    // att = tanh(hidden @ att_w2 + hsW1[b])   M=12800 N=128 Kd=128
    gemm16_kernel<0, 2><<<(BL / 16) * (D_ / 16) / 8, 256, 0, stream>>>(
        hidden, att_w2, w_att, nullptr, w_hsw1, nullptr, nullptr,
        BL, D_, D_, D_, D_, D_);

    beta_kernel<<<BL, 128, 0, stream>>>(w_g, w_att, w2, att_v, mask, w_beta);
    sim_kernel<<<B_, 32, 0, stream>>>(w_beta, mask, w_sims);
    loss_kernel<<<1, B_, 0, stream>>>(w_sims, out_loss);
    select_kernel<<<K_ * B_, D_, 0, stream>>>(w_beta, hidden, w_sel);
    bnorm_kernel<<<NV, D_, 0, stream>>>(emb, w_bn);

    // scores + fused max-over-K: 16 b-tiles * 3125 v-tiles = 50000 waves
    const int NT = (NV + 15) / 16;
    scores_kernel<<<((B_ / 16) * NT + 7) / 8, 256, 0, stream>>>(
        w_sel, w_bn, out_score, out_max);
}